// GraphTrajSTEncoder_67362267070834
// MI455X (gfx1250) — compile-verified
//
#include <hip/hip_runtime.h>

// GraphTrajSTEncoder for MI455X (gfx1250, wave32).
// GEMMs via V_WMMA_F32_16X16X4_F32 (fp32-exact); graph scatter via L2-resident
// f32 atomics (all per-layer node tensors are 25.6MB << 192MB L2).

typedef __attribute__((ext_vector_type(2))) float v2f;
typedef __attribute__((ext_vector_type(8))) float v8f;

#define D_FEAT 128

__device__ __forceinline__ v8f wmma4(v2f a, v2f b, v8f c) {
  return __builtin_amdgcn_wmma_f32_16x16x4_f32(
      /*neg_a=*/false, a, /*neg_b=*/false, b,
      /*c_mod=*/(short)0, c, /*reuse_a=*/false, /*reuse_b=*/false);
}

// ---------------- degree / normalization ----------------

__global__ void k_deg_init(float* __restrict__ deg0, float* __restrict__ deg1, int n) {
  int i = blockIdx.x * blockDim.x + threadIdx.x;
  if (i < n) { deg0[i] = 1.0f; deg1[i] = 1.0f; }   // self-loop contributes 1
}

__global__ void k_deg_count(const int* __restrict__ col, float* __restrict__ deg, int e) {
  int i = blockIdx.x * blockDim.x + threadIdx.x;
  if (i < e) atomicAdd(&deg[col[i]], 1.0f);
}

__global__ void k_deg_to_dis(float* __restrict__ deg, int n) {
  int i = blockIdx.x * blockDim.x + threadIdx.x;
  if (i < n) deg[i] = rsqrtf(deg[i]);              // deg >= 1 always (self loop)
}

// ---------------- fp32 WMMA GEMMs ----------------
// Fragment layouts per CDNA5 ISA 7.12.2 (32-bit, 16x16x4):
//   A 16x4 : lanes 0-15 rows M=0..15 hold {K+0,K+1}; lanes 16-31 hold {K+2,K+3}
//   B 4x16 : cols striped across lanes; VGPR0 rows {K+0|K+2}, VGPR1 rows {K+1|K+3}
//   C/D    : VGPR v -> row m0 + v + 8*(lane>=16), col = lane&15

// C[n x 128] = A[n x 128] @ W[128 x 128].  No bounds checks in the K loop.
__global__ void k_gemm128_wmma(const float* __restrict__ A, const float* __restrict__ W,
                               float* __restrict__ C, int n) {
  const int lane = threadIdx.x & 31;
  const int wave = threadIdx.x >> 5;
  const int mblk = blockIdx.x * (blockDim.x >> 5) + wave;
  if (mblk >= ((n + 15) >> 4)) return;             // wave-uniform exit
  const int m0   = mblk << 4;
  const int half = lane >> 4;
  const int l    = lane & 15;
  const float* __restrict__ arow = A + (size_t)(m0 + l) * 128 + half * 2;

  v8f acc[8] = {};
  for (int k0 = 0; k0 < 128; k0 += 4) {
    const v2f a = *(const v2f*)(arow + k0);        // global_load_b64, 8B aligned
    const int kk = k0 + half * 2;
    const float* __restrict__ w0 = W + (size_t)kk * 128 + l;
#pragma unroll
    for (int t = 0; t < 8; ++t) {
      v2f b;
      b.x = w0[t * 16];
      b.y = w0[t * 16 + 128];
      acc[t] = wmma4(a, b, acc[t]);
    }
  }
#pragma unroll
  for (int t = 0; t < 8; ++t)
#pragma unroll
    for (int v = 0; v < 8; ++v)
      C[(size_t)(m0 + v + half * 8) * D_FEAT + t * 16 + l] = acc[t][v];
}

// C[n x 128] = [X | Pe] @ W,  X: n x 128, Pe: n x peW, W: (128+peW) x 128.
// Phase 1: K in [0,128) from X (guard-free). Phase 2: full 4-steps over Pe
// (guard-free). Phase 3: single guarded tail step (uniform per lane-half).
__global__ void k_gemm_pe_wmma(const float* __restrict__ X, const float* __restrict__ Pe,
                               const int peW, const float* __restrict__ W,
                               float* __restrict__ C, int n) {
  const int lane = threadIdx.x & 31;
  const int wave = threadIdx.x >> 5;
  const int mblk = blockIdx.x * (blockDim.x >> 5) + wave;
  if (mblk >= ((n + 15) >> 4)) return;
  const int m0   = mblk << 4;
  const int half = lane >> 4;
  const int l    = lane & 15;
  const int m    = m0 + l;

  v8f acc[8] = {};

  // ---- phase 1: X part, K = 0..127 ----
  {
    const float* __restrict__ arow = X + (size_t)m * 128 + half * 2;
    for (int k0 = 0; k0 < 128; k0 += 4) {
      const v2f a = *(const v2f*)(arow + k0);
      const int kk = k0 + half * 2;
      const float* __restrict__ w0 = W + (size_t)kk * 128 + l;
#pragma unroll
      for (int t = 0; t < 8; ++t) {
        v2f b;
        b.x = w0[t * 16];
        b.y = w0[t * 16 + 128];
        acc[t] = wmma4(a, b, acc[t]);
      }
    }
  }

  // ---- phase 2: Pe part, full steps (p0+3 < peW) ----
  const int pFull = peW & ~3;                      // 96 for peW=98
  {
    const float* __restrict__ prow = Pe + (size_t)m * peW + half * 2;
    for (int p0 = 0; p0 < pFull; p0 += 4) {
      const v2f a = *(const v2f*)(prow + p0);
      const int kk = 128 + p0 + half * 2;
      const float* __restrict__ w0 = W + (size_t)kk * 128 + l;
#pragma unroll
      for (int t = 0; t < 8; ++t) {
        v2f b;
        b.x = w0[t * 16];
        b.y = w0[t * 16 + 128];
        acc[t] = wmma4(a, b, acc[t]);
      }
    }
  }

  // ---- phase 3: guarded tail (p in [pFull, pFull+4)) ----
  if (pFull < peW) {
    const int p  = pFull + half * 2;
    const int kk = 128 + p;
    const bool g0 = (p     < peW);                 // uniform per lane-half
    const bool g1 = (p + 1 < peW);
    v2f a;
    a.x = g0 ? Pe[(size_t)m * peW + p    ] : 0.0f;
    a.y = g1 ? Pe[(size_t)m * peW + p + 1] : 0.0f;
#pragma unroll
    for (int t = 0; t < 8; ++t) {
      const int col = t * 16 + l;
      v2f b;
      b.x = g0 ? W[(size_t)kk * 128 + col]       : 0.0f;
      b.y = g1 ? W[(size_t)(kk + 1) * 128 + col] : 0.0f;
      acc[t] = wmma4(a, b, acc[t]);
    }
  }

#pragma unroll
  for (int t = 0; t < 8; ++t)
#pragma unroll
    for (int v = 0; v < 8; ++v)
      C[(size_t)(m0 + v + half * 8) * D_FEAT + t * 16 + l] = acc[t][v];
}

// ---------------- aggregation ----------------

// out[i,:] = dis[i]^2 * h1[i,:] + h2[i,:]   (self-loop term; also zero-inits agg)
__global__ void k_selfloop_init(const float* __restrict__ dis,
                                const float* __restrict__ h1,
                                const float* __restrict__ h2,
                                float* __restrict__ out, int n) {
  int i = blockIdx.x * blockDim.x + threadIdx.x;   // over n*32 float4s
  if (i < n * (D_FEAT / 4)) {
    const int node = i >> 5;
    const float s = dis[node];
    const float s2 = s * s;
    const float4 a = ((const float4*)h1)[i];
    const float4 b = ((const float4*)h2)[i];
    float4 o;
    o.x = s2 * a.x + b.x;  o.y = s2 * a.y + b.y;
    o.z = s2 * a.z + b.z;  o.w = s2 * a.w + b.w;
    ((float4*)out)[i] = o;
  }
}

// one wave32 per edge, float4 per lane: out[col,:] += dn*h1[row,:] + en*h2[row,:]
__global__ void k_scatter(const int* __restrict__ row, const int* __restrict__ col,
                          const float* __restrict__ attr, const float* __restrict__ dis,
                          const float* __restrict__ h1, const float* __restrict__ h2,
                          float* __restrict__ out, int e) {
  const int eidx = blockIdx.x * (blockDim.x >> 5) + (threadIdx.x >> 5);
  if (eidx >= e) return;
  const int lane = threadIdx.x & 31;
  const int r = row[eidx];
  const int c = col[eidx];
  const float a = attr[eidx];
  const float en = (a > 0.0f) ? fminf(rsqrtf(a), 1.0f) : 0.0f;
  const float dn = dis[r] * dis[c];
  const float4 h1v = *(const float4*)(h1 + (size_t)r * D_FEAT + lane * 4);  // b128
  const float4 h2v = *(const float4*)(h2 + (size_t)r * D_FEAT + lane * 4);
  float* o = out + (size_t)c * D_FEAT + lane * 4;
  atomicAdd(o + 0, dn * h1v.x + en * h2v.x);
  atomicAdd(o + 1, dn * h1v.y + en * h2v.y);
  atomicAdd(o + 2, dn * h1v.z + en * h2v.z);
  atomicAdd(o + 3, dn * h1v.w + en * h2v.w);
}

__global__ void k_combine_relu(const float* __restrict__ a0, const float* __restrict__ a1,
                               float* __restrict__ out, int total4) {
  int i = blockIdx.x * blockDim.x + threadIdx.x;
  if (i < total4) {
    const float4 a = ((const float4*)a0)[i];
    const float4 b = ((const float4*)a1)[i];
    float4 o;
    o.x = 0.5f * fmaxf(a.x, 0.0f) + 0.5f * fmaxf(b.x, 0.0f);
    o.y = 0.5f * fmaxf(a.y, 0.0f) + 0.5f * fmaxf(b.y, 0.0f);
    o.z = 0.5f * fmaxf(a.z, 0.0f) + 0.5f * fmaxf(b.z, 0.0f);
    o.w = 0.5f * fmaxf(a.w, 0.0f) + 0.5f * fmaxf(b.w, 0.0f);
    ((float4*)out)[i] = o;
  }
}

// ---------------- orchestration ----------------

extern "C" void kernel_launch(void* const* d_in, const int* in_sizes, int n_in,
                              void* d_out, int out_size, void* d_ws, size_t ws_size,
                              hipStream_t stream) {
  const float* x        = (const float*)d_in[0];
  const float* d2an     = (const float*)d_in[1];
  const int*   ei0      = (const int*)  d_in[2];
  const float* ea0      = (const float*)d_in[3];
  const int*   ei1      = (const int*)  d_in[4];
  const float* ea1      = (const float*)d_in[5];
  const float* nodeLin1 = (const float*)d_in[6];
  const float* nodeLin2 = (const float*)d_in[7];
  const float* lin1_1   = (const float*)d_in[8];
  const float* lin2_1   = (const float*)d_in[9];
  const float* lin1_2   = (const float*)d_in[10];
  const float* lin2_2   = (const float*)d_in[11];
  const float* lin1_3   = (const float*)d_in[12];
  const float* lin2_3   = (const float*)d_in[13];
  const float* lin1_4   = (const float*)d_in[14];
  const float* lin2_4   = (const float*)d_in[15];

  const int n  = in_sizes[0] / D_FEAT;          // 50000
  const int pe = in_sizes[1] / n;               // 98
  const int e  = in_sizes[2] / 2;               // 800000
  const int* r0 = ei0;       const int* c0 = ei0 + e;
  const int* r1 = ei1;       const int* c1 = ei1 + e;

  float* ws = (float*)d_ws;
  const size_t ND = (size_t)n * D_FEAT;
  float* dis0 = ws;
  float* dis1 = ws + n;
  float* B1 = ws + 2 * (size_t)n;
  float* B2 = B1 + ND;
  float* B3 = B2 + ND;
  float* B4 = B3 + ND;
  float* B5 = B4 + ND;
  float* B6 = B5 + ND;
  (void)ws_size; (void)n_in;

  const int total4 = n * (D_FEAT / 4);          // float4 count
  const int tpb = 256;
  const int gN     = (n + tpb - 1) / tpb;
  const int gE     = (e + tpb - 1) / tpb;
  const int gTot4  = (total4 + tpb - 1) / tpb;
  const int mtiles = (n + 15) / 16;
  const int gGemm  = (mtiles + 7) / 8;          // 8 waves / block
  const int gScat  = (e + 7) / 8;               // 8 edges / block (wave32 each)

  // --- degrees -> dis (shared by both rounds) ---
  k_deg_init<<<gN, tpb, 0, stream>>>(dis0, dis1, n);
  k_deg_count<<<gE, tpb, 0, stream>>>(c0, dis0, e);
  k_deg_count<<<gE, tpb, 0, stream>>>(c1, dis1, e);
  k_deg_to_dis<<<gN, tpb, 0, stream>>>(dis0, n);
  k_deg_to_dis<<<gN, tpb, 0, stream>>>(dis1, n);

  // --- round 1: PE-fused GEMMs ---
  k_gemm_pe_wmma<<<gGemm, tpb, 0, stream>>>(x, d2an, pe, nodeLin1, B1, n);  // xpe1
  k_gemm_pe_wmma<<<gGemm, tpb, 0, stream>>>(x, d2an, pe, nodeLin2, B2, n);  // xpe2
  k_gemm128_wmma<<<gGemm, tpb, 0, stream>>>(B1, lin1_1, B3, n);             // h1a
  k_gemm128_wmma<<<gGemm, tpb, 0, stream>>>(B1, lin2_1, B4, n);             // h2a
  k_gemm128_wmma<<<gGemm, tpb, 0, stream>>>(B2, lin1_2, B5, n);             // h1b
  k_gemm128_wmma<<<gGemm, tpb, 0, stream>>>(B2, lin2_2, B6, n);             // h2b

  // agg0 -> B1, agg1 -> B2
  k_selfloop_init<<<gTot4, tpb, 0, stream>>>(dis0, B3, B4, B1, n);
  k_scatter<<<gScat, tpb, 0, stream>>>(r0, c0, ea0, dis0, B3, B4, B1, e);
  k_selfloop_init<<<gTot4, tpb, 0, stream>>>(dis1, B5, B6, B2, n);
  k_scatter<<<gScat, tpb, 0, stream>>>(r1, c1, ea1, dis1, B5, B6, B2, e);

  // xm -> B3
  k_combine_relu<<<gTot4, tpb, 0, stream>>>(B1, B2, B3, total4);

  // --- round 2 ---
  k_gemm128_wmma<<<gGemm, tpb, 0, stream>>>(B3, lin1_3, B4, n);             // h1c
  k_gemm128_wmma<<<gGemm, tpb, 0, stream>>>(B3, lin2_3, B5, n);             // h2c
  k_gemm128_wmma<<<gGemm, tpb, 0, stream>>>(B3, lin1_4, B6, n);             // h1d
  k_gemm128_wmma<<<gGemm, tpb, 0, stream>>>(B3, lin2_4, B1, n);             // h2d

  // agg0' -> B2, agg1' -> B3 (xm dead after GEMMs above)
  k_selfloop_init<<<gTot4, tpb, 0, stream>>>(dis0, B4, B5, B2, n);
  k_scatter<<<gScat, tpb, 0, stream>>>(r0, c0, ea0, dis0, B4, B5, B2, e);
  k_selfloop_init<<<gTot4, tpb, 0, stream>>>(dis1, B6, B1, B3, n);
  k_scatter<<<gScat, tpb, 0, stream>>>(r1, c1, ea1, dis1, B6, B1, B3, e);

  k_combine_relu<<<gTot4, tpb, 0, stream>>>(B2, B3, (float*)d_out, total4);
}